// DifferentiableAstar_53987738911392
// MI455X (gfx1250) — compile-verified
//
#include <hip/hip_runtime.h>
#include <math.h>

// Problem constants (match reference)
#define B_ 128
#define H_ 128
#define W_ 128
#define N_ (H_ * W_)          // 16384 cells
#define T_STEPS 204           // int(0.0125 * 128 * 128)
#define NTHREADS 1024         // 32 wave32 waves: fills a WGP
#define CPT (N_ / NTHREADS)   // 16 cells per thread
#define INV_SQRT_W 0.08838834764831845f  // 1/sqrt(128)

// ---- LDS layout (dynamic extern shared; ~216KB, fits CDNA5 320KB/WGP) ----
#define OFF_G      0                        // float g[16384]        64KB
#define OFF_COST   (OFF_G + N_ * 4)         // float cost[16384]     64KB (TDM target)
#define OFF_PARENT (OFF_COST + N_ * 4)      // u16 parent[16384]     32KB
#define OFF_OPEN   (OFF_PARENT + N_ * 2)    // u8 open[16384]        16KB
#define OFF_HIST   (OFF_OPEN + N_)          // u8 hist[16384]        16KB
#define OFF_OBST   (OFF_HIST + N_)          // u8 obst[16384]        16KB (reused as path)
#define OFF_REDV   (OFF_OBST + N_)          // float redv[1024]       4KB
#define OFF_REDI   (OFF_REDV + NTHREADS*4)  // int   redi[1024]       4KB
#define SMEM_BYTES (OFF_REDI + NTHREADS*4)  // 221184 bytes

typedef unsigned int u32x4 __attribute__((ext_vector_type(4)));
typedef int          i32x8 __attribute__((ext_vector_type(8)));
typedef int          i32x4 __attribute__((ext_vector_type(4)));

extern "C" __global__ __launch_bounds__(NTHREADS)
void astar_tdm_kernel(const float* __restrict__ cost,
                      const float* __restrict__ start,
                      const float* __restrict__ goal,
                      const float* __restrict__ obst,
                      float* __restrict__ out)
{
    extern __shared__ char smem[];
    float*          s_g    = (float*)(smem + OFF_G);
    float*          s_cost = (float*)(smem + OFF_COST);
    unsigned short* s_par  = (unsigned short*)(smem + OFF_PARENT);
    unsigned char*  s_open = (unsigned char*)(smem + OFF_OPEN);
    unsigned char*  s_hist = (unsigned char*)(smem + OFF_HIST);
    unsigned char*  s_obst = (unsigned char*)(smem + OFF_OBST); // later reused as path
    float*          s_redv = (float*)(smem + OFF_REDV);
    int*            s_redi = (int*)(smem + OFF_REDI);
    __shared__ int  s_goal_idx;

    const int    t    = threadIdx.x;
    const int    b    = blockIdx.x;
    const size_t base = (size_t)b * N_;

    // ---------- Stage cost map into LDS with the Tensor Data Mover ----------
#if __has_builtin(__builtin_amdgcn_tensor_load_to_lds) && __has_builtin(__builtin_amdgcn_s_wait_tensorcnt)
    if (t < 32) {  // wave 0 issues one TDM op; others sync at the barrier below
        unsigned long long ga = (unsigned long long)(const char*)(cost + base);
        unsigned int lds_byte_addr = (unsigned int)(size_t)s_cost; // low 32b of LDS flat addr
        u32x4 g0;
        g0[0] = 1u;                                   // count=1 (valid), is_restore=0, gather off
        g0[1] = lds_byte_addr;                        // lds_addr[31:0]
        g0[2] = (unsigned int)(ga & 0xFFFFFFFFull);   // global_addr[31:0]
        g0[3] = (unsigned int)((ga >> 32) & 0x01FFFFFFull) | (2u << 30); // addr[56:32] | type=2
        i32x8 g1;
        g1[0] = 0x00020000;                 // workgroup_mask=0, data_size=2 (4 bytes)
        g1[1] = (int)((N_ & 0xFFFF) << 16); // tensor_dim0[15:0] << 16
        g1[2] = (int)(1 << 16);             // tensor_dim0 hi = 0 | tensor_dim1 = 1
        g1[3] = (int)((N_ & 0xFFFF) << 16); // tensor_dim1 hi = 0 | tile_dim0 = 16384
        g1[4] = 1;                          // tile_dim1 = 1, tile_dim2 = 0
        g1[5] = N_;                         // tensor_dim0_stride = 16384 elements
        g1[6] = 0;
        g1[7] = 0;
        i32x4 gz4 = {0, 0, 0, 0};           // groups 2/3 unused (<=2D tensor)
        i32x8 gz8 = {0, 0, 0, 0, 0, 0, 0, 0};
        // clang-23 / therock-10.0 form: 6 args (g0, g1, g2, g3, extra x8 group, cpol)
        __builtin_amdgcn_tensor_load_to_lds(g0, g1, gz4, gz4, gz8, 0);
        __builtin_amdgcn_s_wait_tensorcnt(0);
    }
#else
    #pragma unroll
    for (int k = 0; k < CPT; ++k) { int c = t + k * NTHREADS; s_cost[c] = cost[base + c]; }
#endif

    // ---------- Init state (exactly one goal cell writes s_goal_idx) ----------
    #pragma unroll
    for (int k = 0; k < CPT; ++k) {
        int c = t + k * NTHREADS;
        s_g[c]    = 0.0f;
        s_open[c] = (start[base + c] != 0.0f) ? 1 : 0;
        s_hist[c] = 0;
        s_obst[c] = (obst[base + c] != 0.0f) ? 1 : 0;
        if (goal[base + c] > 0.5f) s_goal_idx = c;
    }
    __syncthreads();
    const int   gidx = s_goal_idx;
    const float grf  = (float)(gidx >> 7);
    const float gcf  = (float)(gidx & 127);
    #pragma unroll
    for (int k = 0; k < CPT; ++k) { int c = t + k * NTHREADS; s_par[c] = (unsigned short)gidx; }
    __syncthreads();

    // ---------- Main sequential A* loop: 204 iterations ----------
    #pragma unroll 1
    for (int it = 0; it < T_STEPS; ++it) {
        // Phase 1: per-thread scan of 16 strided cells; key = exp(-f/sqrt(W))*open
        float bv = -1.0f;
        int   bi = 0;
        #pragma unroll
        for (int k = 0; k < CPT; ++k) {
            int   c   = t + k * NTHREADS;
            float key = 0.0f;
            if (s_open[c]) {
                float dr  = fabsf((float)(c >> 7) - grf);
                float dc  = fabsf((float)(c & 127) - gcf);
                float heu = dr + dc - fminf(dr, dc) + 0.001f * sqrtf(dr * dr + dc * dc);
                float f   = 0.5f * s_g[c] + 0.5f * (heu + s_cost[c]);
                key = expf(-f * INV_SQRT_W);
            }
            if (key > bv || (key == bv && c < bi)) { bv = key; bi = c; }
        }
        s_redv[t] = bv;
        s_redi[t] = bi;
        __syncthreads();
        // Phase 2: tree reduce argmax with jnp.argmax tie rule (lowest index)
        for (int off = NTHREADS >> 1; off > 0; off >>= 1) {
            if (t < off) {
                float ov = s_redv[t + off]; int oi = s_redi[t + off];
                float mv = s_redv[t];       int mi = s_redi[t];
                if (ov > mv || (ov == mv && oi < mi)) { s_redv[t] = ov; s_redi[t] = oi; }
            }
            __syncthreads();
        }
        // Phase 3: O(1) state update by thread 0
        if (t == 0) {
            int s = s_redi[0];
            s_hist[s] = 1;
            if (s != gidx) s_open[s] = 0;   // "unsolved" removal from open set
            float g2v = s_g[s] + s_cost[s];
            int r = s >> 7, c0 = s & 127;
            #pragma unroll
            for (int dr = -1; dr <= 1; ++dr) {
                #pragma unroll
                for (int dc = -1; dc <= 1; ++dc) {
                    if (dr == 0 && dc == 0) continue;
                    int nr = r + dr, nc = c0 + dc;
                    if (nr < 0 || nr >= H_ || nc < 0 || nc >= W_) continue;
                    int n = (nr << 7) | nc;
                    if (!s_obst[n]) continue;
                    // idx = ((1-open)*(1-hist) + open*(g>g2)) * nbr  (in {0,1})
                    bool take = s_open[n] ? (s_g[n] > g2v) : (s_hist[n] == 0);
                    if (take) {
                        s_g[n]    = g2v;
                        s_open[n] = 1;
                        s_par[n]  = (unsigned short)s;
                    }
                }
            }
        }
        __syncthreads();
    }

    // ---------- Backtrack (reuse obstacle buffer as path) ----------
    #pragma unroll
    for (int k = 0; k < CPT; ++k) { int c = t + k * NTHREADS; s_obst[c] = 0; }
    __syncthreads();
    if (t == 0) {
        s_obst[gidx] = 1;                       // path starts as goal one-hot
        int loc = s_par[gidx];
        #pragma unroll 1
        for (int it = 0; it < T_STEPS - 1; ++it) { s_obst[loc] = 1; loc = s_par[loc]; }
    }
    __syncthreads();

    // ---------- Write outputs: hist then paths, concatenated ----------
    float* out_hist = out;
    float* out_path = out + (size_t)B_ * N_;
    #pragma unroll
    for (int k = 0; k < CPT; ++k) {
        int c = t + k * NTHREADS;
        out_hist[base + c] = (float)s_hist[c];
        out_path[base + c] = (float)s_obst[c];
    }
}

extern "C" void kernel_launch(void* const* d_in, const int* in_sizes, int n_in,
                              void* d_out, int out_size, void* d_ws, size_t ws_size,
                              hipStream_t stream) {
    const float* cost  = (const float*)d_in[0];
    const float* start = (const float*)d_in[1];
    const float* goal  = (const float*)d_in[2];
    const float* obst  = (const float*)d_in[3];
    float* out = (float*)d_out;
    astar_tdm_kernel<<<dim3(B_), dim3(NTHREADS), SMEM_BYTES, stream>>>(cost, start, goal, obst, out);
}